// Encoder_7069516169674
// MI455X (gfx1250) — compile-verified
//
#include <hip/hip_runtime.h>

// ---------------------------------------------------------------------------
// 3-layer GraphSAGE (mean agg) + per-channel PReLU, N=50000, E=800000, D=64.
// Sparse scatter is L2-atomic bound; dense part uses V_WMMA_F32_16X16X4_F32.
// Weights staged TRANSPOSED in LDS (stride 68) so each B fragment is one
// bank-conflict-free ds_load_b64 directly into an even-aligned VGPR pair.
// ---------------------------------------------------------------------------

typedef float v2f __attribute__((ext_vector_type(2)));
typedef float v8f __attribute__((ext_vector_type(8)));

#define N_NODES 50000
#define N_EDGES 800000
#define DIM     64
#define BATCH   1024
#define WT_STRIDE 68   // 68 mod 64 == 4 -> conflict-free b64 reads; keeps 8B align

// Zero agg[0:rows*DIM] and cnt[0:rows].
__global__ void zero_kernel(float* __restrict__ agg, float* __restrict__ cnt, int rows) {
    int idx = blockIdx.x * blockDim.x + threadIdx.x;
    if (idx < rows * DIM) agg[idx] = 0.0f;
    if (idx < rows)       cnt[idx] = 0.0f;
}

// One wave per edge. Each lane moves 2 floats of the 64-dim message via
// global_atomic_add_f32; lane 0 increments the in-degree count.
// dstLimit lets layer 2 skip contributions to nodes we never output.
__global__ void scatter_kernel(const float* __restrict__ xin,
                               const long long* __restrict__ src,
                               const long long* __restrict__ dst,
                               float* __restrict__ agg,
                               float* __restrict__ cnt,
                               int nEdges, int dstLimit) {
    int t    = blockIdx.x * blockDim.x + threadIdx.x;
    int e    = t >> 5;
    int lane = t & 31;
    if (e >= nEdges) return;
    int d = (int)dst[e];
    if (d >= dstLimit) return;
    int s = (int)src[e];
    const float2 v = *(const float2*)(xin + (size_t)s * DIM + lane * 2);
    atomicAdd(agg + (size_t)d * DIM + lane * 2 + 0, v.x);
    atomicAdd(agg + (size_t)d * DIM + lane * 2 + 1, v.y);
    if (lane == 0) atomicAdd(cnt + d, 1.0f);
}

// Fused: out = PReLU( (agg/max(cnt,1)) @ Wl + bl + xin @ Wr ).
// 256 threads = 8 waves; each wave computes a 16x64 output tile with
// V_WMMA_F32_16X16X4_F32 (16 k-steps x 4 n-tiles x 2 GEMMs = 128 WMMA/wave).
__global__ void __launch_bounds__(256) sage_gemm_kernel(
    const float* __restrict__ xin, const float* __restrict__ agg,
    const float* __restrict__ cnt,
    const float* __restrict__ Wl, const float* __restrict__ bl,
    const float* __restrict__ Wr, const float* __restrict__ av,
    float* __restrict__ out, int rows)
{
    // Transposed weight tiles: sW?[n * WT_STRIDE + k] = W?[k * DIM + n]
    __shared__ float sWl[DIM * WT_STRIDE];
    __shared__ float sWr[DIM * WT_STRIDE];
    const int tid = threadIdx.x;

    // Stage + transpose both 64x64 weight matrices into LDS.
    // 1024 float4 chunks per matrix / 256 threads = 4 iterations.
    for (int i = tid; i < (DIM * DIM) / 4; i += 256) {
        const int elem = i * 4;            // row-major element index
        const int k  = elem >> 6;          // source row (K)
        const int n0 = elem & 63;          // source col (N), 4 consecutive
        const float4 wl4 = *(const float4*)(Wl + elem);
        const float4 wr4 = *(const float4*)(Wr + elem);
        sWl[(n0 + 0) * WT_STRIDE + k] = wl4.x;
        sWl[(n0 + 1) * WT_STRIDE + k] = wl4.y;
        sWl[(n0 + 2) * WT_STRIDE + k] = wl4.z;
        sWl[(n0 + 3) * WT_STRIDE + k] = wl4.w;
        sWr[(n0 + 0) * WT_STRIDE + k] = wr4.x;
        sWr[(n0 + 1) * WT_STRIDE + k] = wr4.y;
        sWr[(n0 + 2) * WT_STRIDE + k] = wr4.z;
        sWr[(n0 + 3) * WT_STRIDE + k] = wr4.w;
    }
    __syncthreads();

    const int wave = tid >> 5;
    const int lane = tid & 31;
    const int m0   = (blockIdx.x * 8 + wave) * 16;
    if (m0 >= rows) return;  // wave-uniform: EXEC stays all-ones for WMMA

    const int col   = lane & 15;   // A-row within tile / B,C,D column
    const int khalf = lane >> 4;   // which K-pair this lane holds
    const int rowA  = m0 + col;

    const float rc = 1.0f / fmaxf(cnt[rowA], 1.0f);

    // Accumulators for 4 n-tiles, pre-loaded with bias (bias is per-column,
    // and every C/D VGPR in a lane maps to the same column).
    v8f acc[4];
#pragma unroll
    for (int nt = 0; nt < 4; ++nt) {
        float b = bl[nt * 16 + col];
#pragma unroll
        for (int r = 0; r < 8; ++r) acc[nt][r] = b;
    }

    const float* xrow = xin + (size_t)rowA * DIM;
    const float* grow = agg + (size_t)rowA * DIM;

    // Per-lane base pointers into the transposed weights for each n-tile.
    const float* wrp[4];
    const float* wlp[4];
#pragma unroll
    for (int nt = 0; nt < 4; ++nt) {
        const int n = nt * 16 + col;
        wrp[nt] = sWr + n * WT_STRIDE;
        wlp[nt] = sWl + n * WT_STRIDE;
    }

#pragma unroll
    for (int kc = 0; kc < DIM; kc += 4) {
        const int k = kc + khalf * 2;
        // A fragments: lane holds A[M=rowA][K=k, k+1] (one global_load_b64).
        v2f ax = *(const v2f*)(xrow + k);
        v2f ag = *(const v2f*)(grow + k);
        ag[0] *= rc;
        ag[1] *= rc;
#pragma unroll
        for (int nt = 0; nt < 4; ++nt) {
            // B fragments: contiguous {W[k][n], W[k+1][n]} -> one ds_load_b64.
            v2f br  = *(const v2f*)(wrp[nt] + k);
            v2f blv = *(const v2f*)(wlp[nt] + k);
            acc[nt] = __builtin_amdgcn_wmma_f32_16x16x4_f32(
                false, ax, false, br, (short)0, acc[nt], false, false);
            acc[nt] = __builtin_amdgcn_wmma_f32_16x16x4_f32(
                false, ag, false, blv, (short)0, acc[nt], false, false);
        }
    }

    // PReLU + store. D VGPR r -> row m0 + r + khalf*8, column n.
#pragma unroll
    for (int nt = 0; nt < 4; ++nt) {
        const int n     = nt * 16 + col;
        const float slp = av[n];
#pragma unroll
        for (int r = 0; r < 8; ++r) {
            const int row = m0 + r + khalf * 8;
            const float h = acc[nt][r];
            out[(size_t)row * DIM + n] = h > 0.0f ? h : slp * h;
        }
    }
}

extern "C" void kernel_launch(void* const* d_in, const int* in_sizes, int n_in,
                              void* d_out, int out_size, void* d_ws, size_t ws_size,
                              hipStream_t stream) {
    const float*     x   = (const float*)d_in[0];
    const long long* ei  = (const long long*)d_in[1];   // int64 [2, E]
    const long long* src = ei;
    const long long* dst = ei + N_EDGES;

    const float* Wl[3] = { (const float*)d_in[3],  (const float*)d_in[7],  (const float*)d_in[11] };
    const float* bl[3] = { (const float*)d_in[4],  (const float*)d_in[8],  (const float*)d_in[12] };
    const float* Wr[3] = { (const float*)d_in[5],  (const float*)d_in[9],  (const float*)d_in[13] };
    const float* av[3] = { (const float*)d_in[6],  (const float*)d_in[10], (const float*)d_in[14] };

    // Workspace layout (all 16B-aligned given d_ws alignment):
    float* agg = (float*)d_ws;                 // N*64 floats
    float* cnt = agg + (size_t)N_NODES * DIM;  // N floats
    float* h0  = cnt + N_NODES;                // N*64 floats
    float* h1  = h0 + (size_t)N_NODES * DIM;   // N*64 floats

    const int zeroBlocksFull  = (N_NODES * DIM + 255) / 256;
    const int zeroBlocksBatch = (BATCH * DIM + 255) / 256;
    const int scatBlocks      = (N_EDGES * 32 + 255) / 256;
    const int gemmBlocksFull  = ((N_NODES / 16) + 7) / 8;   // 3125 tiles -> 391 blocks
    const int gemmBlocksBatch = ((BATCH / 16) + 7) / 8;     // 64 tiles   -> 8 blocks

    // ---- Layer 0 ----
    zero_kernel<<<zeroBlocksFull, 256, 0, stream>>>(agg, cnt, N_NODES);
    scatter_kernel<<<scatBlocks, 256, 0, stream>>>(x, src, dst, agg, cnt, N_EDGES, N_NODES);
    sage_gemm_kernel<<<gemmBlocksFull, 256, 0, stream>>>(
        x, agg, cnt, Wl[0], bl[0], Wr[0], av[0], h0, N_NODES);

    // ---- Layer 1 ----
    zero_kernel<<<zeroBlocksFull, 256, 0, stream>>>(agg, cnt, N_NODES);
    scatter_kernel<<<scatBlocks, 256, 0, stream>>>(h0, src, dst, agg, cnt, N_EDGES, N_NODES);
    sage_gemm_kernel<<<gemmBlocksFull, 256, 0, stream>>>(
        h0, agg, cnt, Wl[1], bl[1], Wr[1], av[1], h1, N_NODES);

    // ---- Layer 2: only the first BATCH rows are ever read ----
    zero_kernel<<<zeroBlocksBatch, 256, 0, stream>>>(agg, cnt, BATCH);
    scatter_kernel<<<scatBlocks, 256, 0, stream>>>(h1, src, dst, agg, cnt, N_EDGES, BATCH);
    sage_gemm_kernel<<<gemmBlocksBatch, 256, 0, stream>>>(
        h1, agg, cnt, Wl[2], bl[2], Wr[2], av[2], (float*)d_out, BATCH);
}